// PagedAttention_12343736009135
// MI455X (gfx1250) — compile-verified
//
#include <hip/hip_runtime.h>
#include <hip/hip_bf16.h>

typedef __attribute__((ext_vector_type(2))) float v2f;
typedef __attribute__((ext_vector_type(8))) float v8f;

#define NUM_HEADS    8
#define HEAD_DIM     64
#define KV_BLOCK     16
#define NUM_LOGICAL  2048
#define Q_LEN        16
#define HIDDEN       512
#define NCHUNK       128                       // chunks per head (flash-decoding split)
#define CHUNK_BLOCKS (NUM_LOGICAL / NCHUNK)    // 16 KV blocks per chunk (256 kv positions)
#define PART_STRIDE  (16 + 16 + Q_LEN * HEAD_DIM)  // m[16], l[16], o[16][64] floats
#define LOG2E        1.4426950408889634f

// DPP16 reduction step within 16-lane rows (never crosses the half-wave).
#define DPP_STEP_MAX(x, CTRL) \
  x = fmaxf(x, __int_as_float(__builtin_amdgcn_update_dpp(0, __float_as_int(x), CTRL, 0xF, 0xF, false)))
#define DPP_STEP_ADD(x, CTRL) \
  x = x + __int_as_float(__builtin_amdgcn_update_dpp(0, __float_as_int(x), CTRL, 0xF, 0xF, false))

__device__ __forceinline__ float row16_max(float x) {
  DPP_STEP_MAX(x, 0xB1);   // quad_perm xor1
  DPP_STEP_MAX(x, 0x4E);   // quad_perm xor2
  DPP_STEP_MAX(x, 0x141);  // row_half_mirror (xor7)
  DPP_STEP_MAX(x, 0x140);  // row_mirror (xor15)
  return x;
}
__device__ __forceinline__ float row16_sum(float x) {
  DPP_STEP_ADD(x, 0xB1);
  DPP_STEP_ADD(x, 0x4E);
  DPP_STEP_ADD(x, 0x141);
  DPP_STEP_ADD(x, 0x140);
  return x;
}

// ---------------------------------------------------------------------------
// Kernel 1: q = (x @ Wq^T + bq) * rsqrt(D) * log2(e), stored [h][s][d]
// (softmax runs in base-2 throughout)
// ---------------------------------------------------------------------------
__global__ void qproj_kernel(const float* __restrict__ x, const float* __restrict__ Wq,
                             const float* __restrict__ bq, float* __restrict__ qout) {
  int j = blockIdx.x * 64 + threadIdx.x;          // 0..511
  float acc[Q_LEN];
#pragma unroll
  for (int s = 0; s < Q_LEN; ++s) acc[s] = 0.f;
  const float* wrow = Wq + (size_t)j * HIDDEN;
  for (int k = 0; k < HIDDEN; ++k) {
    float w = wrow[k];
#pragma unroll
    for (int s = 0; s < Q_LEN; ++s) acc[s] = fmaf(x[s * HIDDEN + k], w, acc[s]);
  }
  int h = j >> 6, d = j & 63;
  float b = bq[j];
#pragma unroll
  for (int s = 0; s < Q_LEN; ++s)
    qout[(h * Q_LEN + s) * HEAD_DIM + d] = (acc[s] + b) * (0.125f * LOG2E);
}

// ---------------------------------------------------------------------------
// Kernel 2: flash-decoding chunk processor. One wave32 per (head, chunk).
// S = Q K^T via v_wmma_f32_16x16x4_f32; online softmax (DPP max, deferred
// denominator); O += P V via WMMA.
// ---------------------------------------------------------------------------
__global__ __launch_bounds__(128) void attn_chunk_kernel(
    const float* __restrict__ q, const float* __restrict__ kc,
    const float* __restrict__ vc, const long long* __restrict__ bt,
    float* __restrict__ part) {
  __shared__ __align__(16) float plds[4][16][18];   // wave-private P staging, stride 18 keeps b64 alignment
  const int lane = threadIdx.x & 31;
  const int w    = threadIdx.x >> 5;
  const int wid  = blockIdx.x * 4 + w;
  const int h     = wid / NCHUNK;
  const int chunk = wid % NCHUNK;
  const int hf   = lane >> 4;      // half-wave id
  const int l15  = lane & 15;
  const int off  = hf * 2;         // A/B fragment K-offset per ISA layout tables

  // Q fragments: A-layout, lane l15 = row M, halves carry K pairs {0,1}/{2,3}
  v2f qf[16];
  const float* qh = q + (size_t)(h * Q_LEN + l15) * HEAD_DIM;
#pragma unroll
  for (int i = 0; i < 16; ++i) qf[i] = *(const v2f*)(qh + 4 * i + off);

  float mrow[8], lacc[8];          // lacc = per-lane (per-column) partial denominator
#pragma unroll
  for (int r = 0; r < 8; ++r) { mrow[r] = -__builtin_inff(); lacc[r] = 0.f; }
  v8f O0 = {}, O1 = {}, O2 = {}, O3 = {};

  for (int lb = 0; lb < CHUNK_BLOCKS; ++lb) {
    long long p = bt[chunk * CHUNK_BLOCKS + lb];
    const float* K = kc + (((size_t)p * NUM_HEADS + h) << 10);
    const float* V = vc + (((size_t)p * NUM_HEADS + h) << 10);
    if (lb + 1 < CHUNK_BLOCKS) {                  // hide block-table indirection
      long long pn = bt[chunk * CHUNK_BLOCKS + lb + 1];
      const float* Kn = kc + (((size_t)pn * NUM_HEADS + h) << 10);
      const float* Vn = vc + (((size_t)pn * NUM_HEADS + h) << 10);
      __builtin_prefetch(Kn + lane * 32, 0, 0);   // global_prefetch_b8, 128B/lane = 4KB tile
      __builtin_prefetch(Vn + lane * 32, 0, 0);
    }

    // ---- S = Q K^T : 16 x v_wmma_f32_16x16x4_f32, accumulate over D=64 ----
    v8f S = {};
    const float* Kr = K + (l15 << 6) + off;       // B-frag: lane = kv row t
#pragma unroll
    for (int i = 0; i < 16; ++i) {
      v2f b = *(const v2f*)(Kr + 4 * i);
      S = __builtin_amdgcn_wmma_f32_16x16x4_f32(false, qf[i], false, b,
                                                (short)0, S, false, false);
    }

    // ---- online softmax (base 2); C-layout: vgpr r = row r / r+8 per half ----
#pragma unroll
    for (int r = 0; r < 8; ++r) {
      float s = S[r];
      float mx = row16_max(s);                      // DPP, stays in half-wave rows
      float mnew = fmaxf(mrow[r], mx);
      float sc = __builtin_amdgcn_exp2f(mrow[r] - mnew);
      float pv = __builtin_amdgcn_exp2f(s - mnew);
      lacc[r] = fmaf(lacc[r], sc, pv);              // deferred row-sum (per column)
      mrow[r] = mnew;
      S[r] = pv;
      O0[r] *= sc; O1[r] *= sc; O2[r] *= sc; O3[r] *= sc;
    }

    // ---- stage P (C-layout -> LDS row-major) for A-fragment reload.
    // Buffer is wave-private; DS pipe is in-order within a wave, so only a
    // scheduling barrier is needed (no block-wide __syncthreads).
#pragma unroll
    for (int r = 0; r < 8; ++r) plds[w][r + hf * 8][l15] = S[r];
    __builtin_amdgcn_wave_barrier();

    // ---- O += P V : 4 k-steps x 4 col-groups of v_wmma_f32_16x16x4_f32 ----
#pragma unroll
    for (int k = 0; k < 4; ++k) {
      v2f a = *(const v2f*)(&plds[w][l15][4 * k + off]);
      const float* Vr = V + (size_t)(4 * k + off) * HEAD_DIM + l15;
      v2f b;
      b.x = Vr[0];  b.y = Vr[HEAD_DIM];
      O0 = __builtin_amdgcn_wmma_f32_16x16x4_f32(false, a, false, b, (short)0, O0, false, false);
      b.x = Vr[16]; b.y = Vr[HEAD_DIM + 16];
      O1 = __builtin_amdgcn_wmma_f32_16x16x4_f32(false, a, false, b, (short)0, O1, false, false);
      b.x = Vr[32]; b.y = Vr[HEAD_DIM + 32];
      O2 = __builtin_amdgcn_wmma_f32_16x16x4_f32(false, a, false, b, (short)0, O2, false, false);
      b.x = Vr[48]; b.y = Vr[HEAD_DIM + 48];
      O3 = __builtin_amdgcn_wmma_f32_16x16x4_f32(false, a, false, b, (short)0, O3, false, false);
    }
    __builtin_amdgcn_wave_barrier();   // keep next block's P stores behind these reads
  }

  // ---- finalize denominator: single cross-lane sum per row ----
  float lrow[8];
#pragma unroll
  for (int r = 0; r < 8; ++r) lrow[r] = row16_sum(lacc[r]);

  // ---- write partial (m, l, o) for this (head, chunk) ----
  float* pb = part + (size_t)(h * NCHUNK + chunk) * PART_STRIDE;
  if (l15 == 0) {
#pragma unroll
    for (int r = 0; r < 8; ++r) {
      pb[r + hf * 8]      = mrow[r];
      pb[16 + r + hf * 8] = lrow[r];
    }
  }
  float* ob = pb + 32;
#pragma unroll
  for (int r = 0; r < 8; ++r) {
    int row = r + hf * 8;
    ob[row * HEAD_DIM +  0 + l15] = O0[r];
    ob[row * HEAD_DIM + 16 + l15] = O1[r];
    ob[row * HEAD_DIM + 32 + l15] = O2[r];
    ob[row * HEAD_DIM + 48 + l15] = O3[r];
  }
}

// ---------------------------------------------------------------------------
// Kernel 3: log-sum-exp combine of NCHUNK partials per (head, row) (base 2)
// ---------------------------------------------------------------------------
__global__ void combine_kernel(const float* __restrict__ part, float* __restrict__ oat) {
  int h = blockIdx.x >> 4;
  int s = blockIdx.x & 15;
  int d = threadIdx.x;
  const float* pb = part + (size_t)h * NCHUNK * PART_STRIDE;
  float M = -__builtin_inff();
  for (int c = 0; c < NCHUNK; ++c) M = fmaxf(M, pb[(size_t)c * PART_STRIDE + s]);
  float L = 0.f, acc = 0.f;
  for (int c = 0; c < NCHUNK; ++c) {
    const float* q = pb + (size_t)c * PART_STRIDE;
    float wgt = __builtin_amdgcn_exp2f(q[s] - M);
    L  = fmaf(q[16 + s], wgt, L);
    acc = fmaf(wgt, q[32 + s * HEAD_DIM + d], acc);
  }
  oat[s * HIDDEN + h * HEAD_DIM + d] = acc / L;
}

// ---------------------------------------------------------------------------
// Kernel 4: out = o_attn @ Wo^T + bo
// ---------------------------------------------------------------------------
__global__ void oproj_kernel(const float* __restrict__ oat, const float* __restrict__ Wo,
                             const float* __restrict__ bo, float* __restrict__ out) {
  int j = blockIdx.x * 64 + threadIdx.x;          // 0..511
  float acc[Q_LEN];
#pragma unroll
  for (int s = 0; s < Q_LEN; ++s) acc[s] = 0.f;
  const float* wrow = Wo + (size_t)j * HIDDEN;
  for (int k = 0; k < HIDDEN; ++k) {
    float w = wrow[k];
#pragma unroll
    for (int s = 0; s < Q_LEN; ++s) acc[s] = fmaf(oat[s * HIDDEN + k], w, acc[s]);
  }
  float b = bo[j];
#pragma unroll
  for (int s = 0; s < Q_LEN; ++s) out[s * HIDDEN + j] = acc[s] + b;
}

// ---------------------------------------------------------------------------
extern "C" void kernel_launch(void* const* d_in, const int* in_sizes, int n_in,
                              void* d_out, int out_size, void* d_ws, size_t ws_size,
                              hipStream_t stream) {
  const float* hidden = (const float*)d_in[0];
  const float* kcache = (const float*)d_in[1];
  const float* vcache = (const float*)d_in[2];
  const float* Wq     = (const float*)d_in[3];
  const float* bq     = (const float*)d_in[4];
  const float* Wo     = (const float*)d_in[5];
  const float* bo     = (const float*)d_in[6];
  const long long* bt = (const long long*)d_in[7];   // reference dtype: int64

  float* ws   = (float*)d_ws;
  float* q_ws = ws;              // 8*16*64      = 8192 floats
  float* oat  = ws + 8192;       // 16*512       = 8192 floats
  float* part = ws + 16384;      // 8*128*1056   = 1,081,344 floats (~4.3 MB)

  qproj_kernel<<<HIDDEN / 64, 64, 0, stream>>>(hidden, Wq, bq, q_ws);
  attn_chunk_kernel<<<(NUM_HEADS * NCHUNK) / 4, 128, 0, stream>>>(q_ws, kcache, vcache, bt, part);
  combine_kernel<<<NUM_HEADS * Q_LEN, 64, 0, stream>>>(part, oat);
  oproj_kernel<<<HIDDEN / 64, 64, 0, stream>>>(oat, Wo, bo, (float*)d_out);
}